// GCN_53858889891997
// MI455X (gfx1250) — compile-verified
//
#include <hip/hip_runtime.h>

#define N_NODES  100000
#define N_EDGES  1600000
#define N_GRAPHS 4096
#define DIM_IN   11
#define DIM_H    256

typedef float v2f __attribute__((ext_vector_type(2)));
typedef float v8f __attribute__((ext_vector_type(8)));

// ---------------- small helper kernels ----------------

__global__ void fill_kernel(float* __restrict__ p, float v, int n) {
    int i = blockIdx.x * blockDim.x + threadIdx.x;
    if (i < n) p[i] = v;
}

__global__ void degree_kernel(const int* __restrict__ dst, float* __restrict__ deg, int nE) {
    int e = blockIdx.x * blockDim.x + threadIdx.x;
    if (e < nE) atomicAdd(&deg[dst[e]], 1.0f);
}

__global__ void rsqrt_kernel(const float* __restrict__ deg, float* __restrict__ dinv, int n) {
    int i = blockIdx.x * blockDim.x + threadIdx.x;
    if (i < n) dinv[i] = rsqrtf(deg[i]);
}

// Pack W[256,256] into pair-interleaved layout so a WMMA B fragment
// (rows k, k+1 at the same column) is one contiguous 8-byte load:
//   Wp[(j*256 + c)*2 + {0,1}] = W[{2j,2j+1}*256 + c]
__global__ void pack_w_kernel(const float* __restrict__ W, float* __restrict__ Wp) {
    int i = blockIdx.x * blockDim.x + threadIdx.x;     // 0 .. 128*256-1
    int j = i >> 8;                                    // k-pair index
    int c = i & 255;                                   // column
    v2f p = { W[(2 * j) * DIM_H + c], W[(2 * j + 1) * DIM_H + c] };
    *(v2f*)(Wp + (size_t)i * 2) = p;
}

// ---------------- layer-1 GEMM: [N,11] x [11,256] ----------------
// K=11 is too small for WMMA; memory-trivial (x is 4.4 MB). One node per block.
__global__ void gemm_in_kernel(const float* __restrict__ x,
                               const float* __restrict__ W,
                               float* __restrict__ out) {
    __shared__ float xs[DIM_IN];
    const int n = blockIdx.x;
    const int f = threadIdx.x;
    if (f < DIM_IN) xs[f] = x[n * DIM_IN + f];
    __syncthreads();
    float acc = 0.0f;
#pragma unroll
    for (int k = 0; k < DIM_IN; ++k)
        acc = fmaf(xs[k], W[k * DIM_H + f], acc);
    out[(size_t)n * DIM_H + f] = acc;
}

// ---------------- WMMA fp32 GEMM: [N,256] x [256,256] ----------------
// V_WMMA_F32_16X16X4_F32, wave32. Each wave owns a 16x64 output strip:
// 4 v8f accumulators reuse one A fragment per K-step (64 K-steps).
// A frag (16x4 f32): lanes 0-15 hold row M=lane, K={k0,k0+1}; lanes 16-31 K={k0+2,k0+3}.
// B frag comes from the pair-packed Wp (one b64 load per fragment).
// C/D (16x16 f32): VGPR v -> M = half*8+v, N = lane&15.
__global__ void gemm_wmma_kernel(const float* __restrict__ A,    // [N_NODES, 256]
                                 const float* __restrict__ Wp,   // packed [128][256][2]
                                 float* __restrict__ out) {      // [N_NODES, 256]
    const int lane  = threadIdx.x & 31;
    const int wave  = threadIdx.x >> 5;              // 0..7
    const int task  = blockIdx.x * 8 + wave;         // 6250*4 = 25000 tasks
    const int mTile = task >> 2;                     // 0..6249
    const int nGrp  = task & 3;                      // 64-column group
    const int half  = lane >> 4;
    const int l16   = lane & 15;

    const float* __restrict__ arow = A + (size_t)(mTile * 16 + l16) * DIM_H;
    const int c0 = nGrp * 64 + l16;

    v8f acc0 = {}, acc1 = {}, acc2 = {}, acc3 = {};

#pragma unroll 4
    for (int k0 = 0; k0 < DIM_H; k0 += 4) {
        const int k = k0 + half * 2;
        const int j = (k0 >> 1) + half;              // k-pair index
        v2f a = *(const v2f*)(arow + k);             // global_load_b64
        const float* __restrict__ wj = Wp + ((size_t)j * DIM_H) * 2;
        v2f b0 = *(const v2f*)(wj + (c0 +  0) * 2);  // global_load_b64 each
        v2f b1 = *(const v2f*)(wj + (c0 + 16) * 2);
        v2f b2 = *(const v2f*)(wj + (c0 + 32) * 2);
        v2f b3 = *(const v2f*)(wj + (c0 + 48) * 2);
        acc0 = __builtin_amdgcn_wmma_f32_16x16x4_f32(false, a, false, b0, (short)0, acc0, false, false);
        acc1 = __builtin_amdgcn_wmma_f32_16x16x4_f32(false, a, false, b1, (short)0, acc1, false, false);
        acc2 = __builtin_amdgcn_wmma_f32_16x16x4_f32(false, a, false, b2, (short)0, acc2, false, false);
        acc3 = __builtin_amdgcn_wmma_f32_16x16x4_f32(false, a, false, b3, (short)0, acc3, false, false);
    }

#pragma unroll
    for (int v = 0; v < 8; ++v) {
        float* orow = out + (size_t)(mTile * 16 + half * 8 + v) * DIM_H;
        orow[c0 +  0] = acc0[v];
        orow[c0 + 16] = acc1[v];
        orow[c0 + 32] = acc2[v];
        orow[c0 + 48] = acc3[v];
    }
}

// ---------------- edge message + scatter-add ----------------
// 4 edges per 256-thread block, 64 threads/edge, float4 gathers (b128).
// h (102.4 MB) and agg are L2-resident (192 MB L2), so gathers + f32
// atomics stay on-chip; atomic count (E*F) is irreducible for this pattern.
__global__ void edge_kernel(const int* __restrict__ src, const int* __restrict__ dst,
                            const float* __restrict__ dinv,
                            const float* __restrict__ h, float* __restrict__ agg) {
    const int e = blockIdx.x * 4 + (threadIdx.x >> 6);
    const int t = threadIdx.x & 63;
    const int s = src[e];
    const int d = dst[e];
    const float coef = dinv[s] * dinv[d];
    const float4 hv = *(const float4*)(h + (size_t)s * DIM_H + t * 4);
    float* ap = agg + (size_t)d * DIM_H + t * 4;
    atomicAdd(ap + 0, hv.x * coef);
    atomicAdd(ap + 1, hv.y * coef);
    atomicAdd(ap + 2, hv.z * coef);
    atomicAdd(ap + 3, hv.w * coef);
}

// ---------------- combine: agg + h*dinv^2 + bias, optional ReLU ----------------
__global__ void combine_kernel(const float4* __restrict__ agg, const float4* __restrict__ t,
                               const float* __restrict__ dinv, const float* __restrict__ bias,
                               float4* __restrict__ out, int relu) {
    const size_t i4 = (size_t)blockIdx.x * blockDim.x + threadIdx.x;
    if (i4 >= (size_t)N_NODES * (DIM_H / 4)) return;
    const int n  = (int)(i4 >> 6);
    const int f4 = (int)(i4 & 63) * 4;
    const float di = dinv[n];
    const float di2 = di * di;
    const float4 a  = agg[i4];
    const float4 tv = t[i4];
    const float4 bv = *(const float4*)(bias + f4);
    float4 v;
    v.x = a.x + tv.x * di2 + bv.x;
    v.y = a.y + tv.y * di2 + bv.y;
    v.z = a.z + tv.z * di2 + bv.z;
    v.w = a.w + tv.w * di2 + bv.w;
    if (relu) {
        v.x = fmaxf(v.x, 0.0f); v.y = fmaxf(v.y, 0.0f);
        v.z = fmaxf(v.z, 0.0f); v.w = fmaxf(v.w, 0.0f);
    }
    out[i4] = v;
}

// ---------------- global mean pool (sums + counts) ----------------
__global__ void pool_kernel(const float* __restrict__ h, const int* __restrict__ batch,
                            float* __restrict__ pooled, float* __restrict__ cnt) {
    const int n = blockIdx.x;
    const int t = threadIdx.x;            // 0..63
    const int g = batch[n];
    const float4 hv = *(const float4*)(h + (size_t)n * DIM_H + t * 4);
    float* pp = pooled + (size_t)g * DIM_H + t * 4;
    atomicAdd(pp + 0, hv.x);
    atomicAdd(pp + 1, hv.y);
    atomicAdd(pp + 2, hv.z);
    atomicAdd(pp + 3, hv.w);
    if (t == 0) atomicAdd(&cnt[g], 1.0f);
}

// ---------------- readout: (pooled/cnt) . Wl + bl ----------------
__global__ void readout_kernel(const float* __restrict__ pooled, const float* __restrict__ cnt,
                               const float* __restrict__ Wl, const float* __restrict__ bl,
                               float* __restrict__ out) {
    __shared__ float red[DIM_H];
    const int g = blockIdx.x;
    const int t = threadIdx.x;
    red[t] = pooled[(size_t)g * DIM_H + t] * Wl[t];
    __syncthreads();
#pragma unroll
    for (int s = DIM_H / 2; s > 0; s >>= 1) {
        if (t < s) red[t] += red[t + s];
        __syncthreads();
    }
    if (t == 0) out[g] = red[0] / fmaxf(cnt[g], 1.0f) + bl[0];
}

// ---------------- orchestration ----------------

extern "C" void kernel_launch(void* const* d_in, const int* in_sizes, int n_in,
                              void* d_out, int out_size, void* d_ws, size_t ws_size,
                              hipStream_t stream) {
    (void)in_sizes; (void)n_in; (void)out_size; (void)ws_size;

    const float* x     = (const float*)d_in[0];
    const int*   edge  = (const int*)d_in[1];
    const int*   src   = edge;
    const int*   dst   = edge + N_EDGES;
    const int*   batch = (const int*)d_in[2];
    const float* W1 = (const float*)d_in[3];  const float* b1 = (const float*)d_in[4];
    const float* W2 = (const float*)d_in[5];  const float* b2 = (const float*)d_in[6];
    const float* W3 = (const float*)d_in[7];  const float* b3 = (const float*)d_in[8];
    const float* Wl = (const float*)d_in[9];  const float* bl = (const float*)d_in[10];
    float* out = (float*)d_out;

    const size_t HN = (size_t)N_NODES * DIM_H;
    float* bufT   = (float*)d_ws;            // GEMM output t = h_in @ W
    float* bufG   = bufT + HN;               // agg (scatter target)
    float* bufH   = bufG + HN;               // layer activation
    float* deg    = bufH + HN;
    float* dinv   = deg + N_NODES;
    float* pooled = dinv + N_NODES;
    float* cnt    = pooled + (size_t)N_GRAPHS * DIM_H;
    float* wpack  = cnt + N_GRAPHS;          // pair-packed W (64K floats)

    const int c4Blocks   = (int)((HN / 4 + 255) / 256);
    const int edgeBlocks = N_EDGES / 4;
    const int packBlocks = (DIM_H / 2) * DIM_H / 256;   // 128*256/256

    // degrees & symmetric normalization
    fill_kernel<<<(N_NODES + 255) / 256, 256, 0, stream>>>(deg, 1.0f, N_NODES); // +1 self loop
    degree_kernel<<<(N_EDGES + 255) / 256, 256, 0, stream>>>(dst, deg, N_EDGES);
    rsqrt_kernel<<<(N_NODES + 255) / 256, 256, 0, stream>>>(deg, dinv, N_NODES);

    // ---- layer 1 (K=11 GEMM, VALU) ----
    gemm_in_kernel<<<N_NODES, DIM_H, 0, stream>>>(x, W1, bufT);
    hipMemsetAsync(bufG, 0, HN * sizeof(float), stream);
    edge_kernel<<<edgeBlocks, 256, 0, stream>>>(src, dst, dinv, bufT, bufG);
    combine_kernel<<<c4Blocks, 256, 0, stream>>>((const float4*)bufG, (const float4*)bufT,
                                                 dinv, b1, (float4*)bufH, 1);

    // ---- layer 2 (WMMA fp32 GEMM) ----
    pack_w_kernel<<<packBlocks, 256, 0, stream>>>(W2, wpack);
    gemm_wmma_kernel<<<25000 / 8, 256, 0, stream>>>(bufH, wpack, bufT);
    hipMemsetAsync(bufG, 0, HN * sizeof(float), stream);
    edge_kernel<<<edgeBlocks, 256, 0, stream>>>(src, dst, dinv, bufT, bufG);
    combine_kernel<<<c4Blocks, 256, 0, stream>>>((const float4*)bufG, (const float4*)bufT,
                                                 dinv, b2, (float4*)bufH, 1);

    // ---- layer 3 (WMMA fp32 GEMM, no ReLU) ----
    pack_w_kernel<<<packBlocks, 256, 0, stream>>>(W3, wpack);
    gemm_wmma_kernel<<<25000 / 8, 256, 0, stream>>>(bufH, wpack, bufT);
    hipMemsetAsync(bufG, 0, HN * sizeof(float), stream);
    edge_kernel<<<edgeBlocks, 256, 0, stream>>>(src, dst, dinv, bufT, bufG);
    combine_kernel<<<c4Blocks, 256, 0, stream>>>((const float4*)bufG, (const float4*)bufT,
                                                 dinv, b3, (float4*)bufH, 0);

    // ---- pooling + readout ----
    hipMemsetAsync(pooled, 0, (size_t)N_GRAPHS * DIM_H * sizeof(float), stream);
    hipMemsetAsync(cnt, 0, (size_t)N_GRAPHS * sizeof(float), stream);
    pool_kernel<<<N_NODES, 64, 0, stream>>>(bufH, batch, pooled, cnt);
    readout_kernel<<<N_GRAPHS, DIM_H, 0, stream>>>(pooled, cnt, Wl, bl, out);
}